// FujiMHC_71159018160274
// MI455X (gfx1250) — compile-verified
//
#include <hip/hip_runtime.h>

// ---------------- CDNA5 (gfx1250) FujiMHC kernel ----------------
// wave32, WMMA f32 16x16x4 for the per-token [16 x 8192] @ [8192 x 32] matvec.

typedef float v2f __attribute__((ext_vector_type(2)));
typedef float v4f __attribute__((ext_vector_type(4)));
typedef float v8f __attribute__((ext_vector_type(8)));

#define N_HEADS     4
#define D_DIM       2048
#define ND          8192          // n * D
#define NPERM       24
#define NALPHA      28            // n + n!
#define NCOL        32            // 28 alpha cols + 4 beta cols, packed
#define TOK_PER_BLK 16
#define THREADS     256
#define WAVES       8
#define K_PER_WAVE  (ND / WAVES)  // 1024
#define CHUNKS      (K_PER_WAVE / 4)

// itertools.permutations(range(4)) in lexicographic order: perm[r][i] = j
__device__ static const int c_perm[NPERM][4] = {
    {0,1,2,3},{0,1,3,2},{0,2,1,3},{0,2,3,1},{0,3,1,2},{0,3,2,1},
    {1,0,2,3},{1,0,3,2},{1,2,0,3},{1,2,3,0},{1,3,0,2},{1,3,2,0},
    {2,0,1,3},{2,0,3,1},{2,1,0,3},{2,1,3,0},{2,3,0,1},{2,3,1,0},
    {3,0,1,2},{3,0,2,1},{3,1,0,2},{3,1,2,0},{3,2,0,1},{3,2,1,0}
};

// Pack [8192x28] alpha_fn and [8192x4] beta_fn into one [4096][32] float2
// buffer (pairs of consecutive k rows interleaved, matching the WMMA B
// fragment layout), with (1 + norm_weight[k]) folded in.
__global__ void pack_w_kernel(const float* __restrict__ nw,
                              const float* __restrict__ afn,
                              const float* __restrict__ bfn,
                              float2* __restrict__ wpk)
{
    int idx = blockIdx.x * blockDim.x + threadIdx.x;   // [(ND/2)*NCOL)
    if (idx >= (ND / 2) * NCOL) return;
    int kk = idx >> 5;          // k pair index
    int j  = idx & 31;
    int k0 = kk * 2, k1 = k0 + 1;
    float g0 = 1.0f + nw[k0];
    float g1 = 1.0f + nw[k1];
    float v0 = (j < NALPHA) ? afn[k0 * NALPHA + j] : bfn[k0 * 4 + (j - NALPHA)];
    float v1 = (j < NALPHA) ? afn[k1 * NALPHA + j] : bfn[k1 * 4 + (j - NALPHA)];
    float2 o; o.x = v0 * g0; o.y = v1 * g1;
    wpk[idx] = o;
}

__device__ __forceinline__ float sigmoidf_(float x) {
    return 1.0f / (1.0f + __expf(-x));
}

__global__ __launch_bounds__(THREADS)
void fuji_mhc_kernel(const float* __restrict__ X,        // [4, BS, D]
                     const float* __restrict__ x_out,    // [BS, D]
                     const float* __restrict__ s_alpha_g,// [28]
                     const float* __restrict__ pre_scale,// [1]
                     const float* __restrict__ res_scale,// [1]
                     const float* __restrict__ s_beta_g, // [4]
                     const float* __restrict__ h_post,   // [1]
                     const float2* __restrict__ wpk,     // [(ND/2)*NCOL]
                     float* __restrict__ out_branch,     // [BS*D]
                     float* __restrict__ out_xnew,       // [4*BS*D]
                     int BS)
{
    __shared__ float s_wc[WAVES * TOK_PER_BLK * NCOL];   // 16 KB
    __shared__ float s_ss[WAVES * 32];
    __shared__ float s_inv[TOK_PER_BLK];
    __shared__ float s_H[TOK_PER_BLK][16];
    __shared__ float s_al[TOK_PER_BLK][4];
    __shared__ float s_bt[TOK_PER_BLK][4];

    const int tid    = threadIdx.x;
    const int w      = tid >> 5;
    const int lane   = tid & 31;
    const int lane16 = lane & 15;
    const int h      = lane >> 4;          // 0: K={0,1}, 1: K={2,3}
    const int t0     = blockIdx.x * TOK_PER_BLK;
    const long NBSD  = (long)BS * D_DIM;

    // ---------------- pass 1: WMMA matvec + sum-of-squares ----------------
    const int kbase = w * K_PER_WAVE;
    const int n_idx = kbase >> 11;             // which of the 4 X rows
    const int dbase = kbase & (D_DIM - 1);
    const float* rowp = X + (long)n_idx * NBSD
                          + (long)(t0 + lane16) * D_DIM + dbase + 2 * h;
    const float2* wp  = wpk + ((kbase >> 1) + h) * NCOL + lane16;

    v8f acc0 = {};   // C tile: tokens x j[0..15]
    v8f acc1 = {};   // C tile: tokens x j[16..31]
    float ss = 0.0f;

    #pragma unroll 4
    for (int kk = 0; kk < CHUNKS; ++kk) {
        v2f a  = *(const v2f*)(rowp + 4 * kk);               // A frag (16x4 f32)
        v2f b0 = *(const v2f*)(wp + (size_t)2 * kk * NCOL);  // B frag j0=0
        v2f b1 = *(const v2f*)(wp + (size_t)2 * kk * NCOL + 16);
        acc0 = __builtin_amdgcn_wmma_f32_16x16x4_f32(
                   false, a, false, b0, (short)0, acc0, false, false);
        acc1 = __builtin_amdgcn_wmma_f32_16x16x4_f32(
                   false, a, false, b1, (short)0, acc1, false, false);
        ss += a[0] * a[0] + a[1] * a[1];
    }

    // spill per-wave partials: lane l vgpr v holds (M = v + 8*h, N = lane16)
    {
        float* dst = s_wc + w * (TOK_PER_BLK * NCOL);
        #pragma unroll
        for (int v = 0; v < 8; ++v) {
            int m = v + 8 * h;
            dst[m * NCOL + lane16]      = acc0[v];
            dst[m * NCOL + 16 + lane16] = acc1[v];
        }
        s_ss[w * 32 + lane] = ss;
    }
    __syncthreads();

    // ---------------- cross-wave reduction ----------------
    for (int e = tid; e < TOK_PER_BLK * NCOL; e += THREADS) {
        float r = 0.0f;
        #pragma unroll
        for (int ww = 0; ww < WAVES; ++ww)
            r += s_wc[ww * (TOK_PER_BLK * NCOL) + e];
        s_wc[e] = r;   // each entry owned by exactly one thread -> safe
    }
    if (tid < TOK_PER_BLK) {
        float r = 0.0f;
        #pragma unroll
        for (int ww = 0; ww < WAVES; ++ww)
            r += s_ss[ww * 32 + tid] + s_ss[ww * 32 + 16 + tid];
        s_inv[tid] = rsqrtf(r * (1.0f / (float)ND) + 1e-6f);
    }
    __syncthreads();

    // ---------------- per-token scalars (16 threads) ----------------
    if (tid < TOK_PER_BLK) {
        const float inv = s_inv[tid];
        float wc[NCOL];
        #pragma unroll
        for (int j = 0; j < NCOL; ++j) wc[j] = s_wc[tid * NCOL + j] * inv;

        const float ps = pre_scale[0], rs = res_scale[0], hp = h_post[0];

        #pragma unroll
        for (int i = 0; i < 4; ++i)
            s_al[tid][i] = sigmoidf_(ps * wc[i] + s_alpha_g[i]);

        float lg[NPERM];
        float mx = -3.0e38f;
        #pragma unroll
        for (int r = 0; r < NPERM; ++r) {
            lg[r] = rs * wc[4 + r] + s_alpha_g[4 + r];
            mx = fmaxf(mx, lg[r]);
        }
        float sum = 0.0f;
        #pragma unroll
        for (int r = 0; r < NPERM; ++r) { lg[r] = __expf(lg[r] - mx); sum += lg[r]; }
        const float isum = 1.0f / sum;

        float H[16];
        #pragma unroll
        for (int q = 0; q < 16; ++q) H[q] = 0.0f;
        #pragma unroll
        for (int r = 0; r < NPERM; ++r) {
            float rc = lg[r] * isum;
            #pragma unroll
            for (int i = 0; i < 4; ++i) H[i * 4 + c_perm[r][i]] += rc;
        }
        #pragma unroll
        for (int q = 0; q < 16; ++q) s_H[tid][q] = H[q];

        #pragma unroll
        for (int i = 0; i < 4; ++i)
            s_bt[tid][i] = 2.0f * sigmoidf_(hp * wc[NALPHA + i] + s_beta_g[i]);
    }
    __syncthreads();

    // ---------------- pass 2: outputs (float4 coalesced; L2-hot X) --------
    const int JOBS = TOK_PER_BLK * (D_DIM / 4);   // 8192
    for (int job = tid; job < JOBS; job += THREADS) {
        const int tl = job >> 9;          // local token
        const int dc = job & 511;         // float4 chunk in D
        const long base = (long)(t0 + tl) * D_DIM + (long)dc * 4;

        v4f x0 = *(const v4f*)(X + base);
        v4f x1 = *(const v4f*)(X + NBSD + base);
        v4f x2 = *(const v4f*)(X + 2 * NBSD + base);
        v4f x3 = *(const v4f*)(X + 3 * NBSD + base);
        v4f xo = *(const v4f*)(x_out + base);

        v4f bi = s_al[tl][0] * x0 + s_al[tl][1] * x1
               + s_al[tl][2] * x2 + s_al[tl][3] * x3;
        *(v4f*)(out_branch + base) = bi;

        #pragma unroll
        for (int i = 0; i < 4; ++i) {
            v4f r = s_bt[tl][i] * xo
                  + s_H[tl][i * 4 + 0] * x0 + s_H[tl][i * 4 + 1] * x1
                  + s_H[tl][i * 4 + 2] * x2 + s_H[tl][i * 4 + 3] * x3;
            *(v4f*)(out_xnew + (long)i * NBSD + base) = r;
        }
    }
}

extern "C" void kernel_launch(void* const* d_in, const int* in_sizes, int n_in,
                              void* d_out, int out_size, void* d_ws, size_t ws_size,
                              hipStream_t stream) {
    const float* X      = (const float*)d_in[0];  // [4,B,S,D]
    const float* x_out  = (const float*)d_in[1];  // [B,S,D]
    const float* nw     = (const float*)d_in[2];  // [8192]
    const float* salpha = (const float*)d_in[3];  // [28]
    const float* afn    = (const float*)d_in[4];  // [8192,28]
    const float* ps     = (const float*)d_in[5];  // [1]
    const float* rs     = (const float*)d_in[6];  // [1]
    const float* sbeta  = (const float*)d_in[7];  // [4]
    const float* bfn    = (const float*)d_in[8];  // [8192,4]
    const float* hp     = (const float*)d_in[9];  // [1]

    float* out = (float*)d_out;
    const int BS = in_sizes[1] / D_DIM;           // B*S tokens (4096)

    // scratch: packed weights, (ND/2)*NCOL float2 = 1 MB
    float2* wpk = (float2*)d_ws;
    const int packN = (ND / 2) * NCOL;
    pack_w_kernel<<<(packN + 255) / 256, 256, 0, stream>>>(nw, afn, bfn, wpk);

    float* out_branch = out;                          // [BS*D]
    float* out_xnew   = out + (long)BS * D_DIM;       // [4*BS*D]
    fuji_mhc_kernel<<<BS / TOK_PER_BLK, THREADS, 0, stream>>>(
        X, x_out, salpha, ps, rs, sbeta, hp, wpk, out_branch, out_xnew, BS);
}